// GNN_71502615544305
// MI455X (gfx1250) — compile-verified
//
#include <hip/hip_runtime.h>
#include <math.h>

#define N_NODES 10000
#define N_EDGES 160000
#define EP      (N_EDGES + N_NODES)   // edges + self loops
#define BATCH   4
#define HID     128
#define HEADS   4
#define HH      (HEADS * HID)         // 512
#define NEG_SLOPE 0.2f
#define LN_EPS    1e-5f

typedef float v2f __attribute__((ext_vector_type(2)));
typedef float v8f __attribute__((ext_vector_type(8)));

__device__ __forceinline__ float waveReduceSum(float v) {
    #pragma unroll
    for (int off = 16; off > 0; off >>= 1)
        v += __shfl_xor(v, off, 32);
    return v;
}

__device__ __forceinline__ void atomicMaxF(float* addr, float val) {
    // monotonic-bits trick: signed max for >=0, unsigned min for <0
    if (val >= 0.0f) atomicMax((int*)addr, __float_as_int(val));
    else             atomicMin((unsigned int*)addr, __float_as_uint(val));
}

// -------------------- GEMM: h[b] = x[b] (N x 128) @ W (128 x 512) ----------
// grid: (N/16, BATCH), block 256 (8 waves). Wave w computes cols [w*64, w*64+64).
__global__ void gemm_wmma_f32(const float* __restrict__ X,
                              const float* __restrict__ W,
                              float* __restrict__ H) {
    __shared__ float At[16 * HID];   // 8 KB A tile
    const int b  = blockIdx.y;
    const int m0 = blockIdx.x * 16;
    const float* xg = X + ((size_t)b * N_NODES + m0) * HID;

    const int t = threadIdx.x;
    ((float4*)At)[t]       = ((const float4*)xg)[t];
    ((float4*)At)[t + 256] = ((const float4*)xg)[t + 256];
    __syncthreads();

    const int wave = t >> 5;
    const int lane = t & 31;
    const int mr   = lane & 15;          // M row within tile
    const int kb   = (lane >> 4) << 1;   // K base: 0 (lanes 0-15) / 2 (lanes 16-31)
    const int rowo = (lane >> 4) << 3;   // D row offset: 0 / 8

    // Preload A fragments: lane holds A[mr][4*kk+kb], A[mr][4*kk+kb+1]
    v2f a[32];
    #pragma unroll
    for (int kk = 0; kk < 32; ++kk)
        a[kk] = *(const v2f*)&At[mr * HID + kk * 4 + kb];

    float* hp = H + ((size_t)b * N_NODES + m0) * HH;

    for (int nt = 0; nt < 4; ++nt) {
        const int n = wave * 64 + nt * 16 + mr;   // output column
        v8f c = {0.f, 0.f, 0.f, 0.f, 0.f, 0.f, 0.f, 0.f};
        #pragma unroll
        for (int kk = 0; kk < 32; ++kk) {
            const int k = kk * 4 + kb;
            v2f bb;
            bb.x = W[(size_t)k * HH + n];
            bb.y = W[(size_t)(k + 1) * HH + n];
            c = __builtin_amdgcn_wmma_f32_16x16x4_f32(
                    false, a[kk], false, bb, (short)0, c, false, false);
        }
        #pragma unroll
        for (int r = 0; r < 8; ++r)
            hp[(size_t)(r + rowo) * HH + n] = c[r];
    }
}

// ---------- per-(node,head) attention logits e_src/e_dst ----------
// wave per (node, head). grid: (ceil(N*HEADS/8), BATCH), block 256.
__global__ void att_e_kernel(const float* __restrict__ H,
                             const float* __restrict__ As,
                             const float* __restrict__ Ad,
                             float* __restrict__ esrc,
                             float* __restrict__ edst) {
    const int b    = blockIdx.y;
    const int wave = threadIdx.x >> 5;
    const int lane = threadIdx.x & 31;
    const int pair = blockIdx.x * 8 + wave;
    if (pair >= N_NODES * HEADS) return;
    const int n  = pair >> 2;
    const int hd = pair & 3;

    const float4 hv = ((const float4*)(H + ((size_t)b * N_NODES + n) * HH + hd * HID))[lane];
    const float4 as = ((const float4*)(As + hd * HID))[lane];
    const float4 ad = ((const float4*)(Ad + hd * HID))[lane];
    float es = hv.x * as.x + hv.y * as.y + hv.z * as.z + hv.w * as.w;
    float ed = hv.x * ad.x + hv.y * ad.y + hv.z * ad.z + hv.w * ad.w;
    es = waveReduceSum(es);
    ed = waveReduceSum(ed);
    if (lane == 0) {
        esrc[((size_t)b * N_NODES + n) * HEADS + hd] = es;
        edst[((size_t)b * N_NODES + n) * HEADS + hd] = ed;
    }
}

// ---------- init: mmax=-inf, ssum=0, agg=0 ----------
__global__ void init_kernel(float* __restrict__ mmax,
                            float* __restrict__ ssum,
                            float* __restrict__ agg) {
    const int b = blockIdx.y;
    const int n = blockIdx.x * 256 + threadIdx.x;
    if (n >= N_NODES) return;
    const float ninf = __int_as_float(0xff800000u);
    ((float4*)mmax)[(size_t)b * N_NODES + n] = make_float4(ninf, ninf, ninf, ninf);
    ((float4*)ssum)[(size_t)b * N_NODES + n] = make_float4(0.f, 0.f, 0.f, 0.f);
    float4* ag = (float4*)(agg + ((size_t)b * N_NODES + n) * HID);
    #pragma unroll
    for (int i = 0; i < HID / 4; ++i)
        ag[i] = make_float4(0.f, 0.f, 0.f, 0.f);
}

// ---------- edge pass 1: leaky-relu logits + segment max ----------
__global__ void edge_max_kernel(const int* __restrict__ srcI,
                                const int* __restrict__ tgtI,
                                const float* __restrict__ esrc,
                                const float* __restrict__ edst,
                                float* __restrict__ ebuf,
                                float* __restrict__ mmax) {
    const int b = blockIdx.y;
    const int e = blockIdx.x * 256 + threadIdx.x;
    if (e >= EP) return;
    int s, tg;
    if (e < N_EDGES) {
        s  = srcI[(size_t)b * N_EDGES + e];
        tg = tgtI[(size_t)b * N_EDGES + e];
    } else {
        s = tg = e - N_EDGES;   // self loop
    }
    const float4 es = ((const float4*)esrc)[(size_t)b * N_NODES + s];
    const float4 ed = ((const float4*)edst)[(size_t)b * N_NODES + tg];
    float v[4] = {es.x + ed.x, es.y + ed.y, es.z + ed.z, es.w + ed.w};
    #pragma unroll
    for (int h = 0; h < 4; ++h)
        v[h] = v[h] > 0.f ? v[h] : NEG_SLOPE * v[h];
    ((float4*)ebuf)[(size_t)b * EP + e] = make_float4(v[0], v[1], v[2], v[3]);
    float* mp = mmax + ((size_t)b * N_NODES + tg) * HEADS;
    #pragma unroll
    for (int h = 0; h < 4; ++h)
        atomicMaxF(mp + h, v[h]);
}

// ---------- edge pass 2: exp(e - max) + segment sum ----------
__global__ void edge_sum_kernel(const int* __restrict__ tgtI,
                                const float* __restrict__ mmax,
                                float* __restrict__ ebuf,
                                float* __restrict__ ssum) {
    const int b = blockIdx.y;
    const int e = blockIdx.x * 256 + threadIdx.x;
    if (e >= EP) return;
    const int tg = (e < N_EDGES) ? tgtI[(size_t)b * N_EDGES + e] : (e - N_EDGES);
    float4 v = ((const float4*)ebuf)[(size_t)b * EP + e];
    const float4 m = ((const float4*)mmax)[(size_t)b * N_NODES + tg];
    float ex[4] = {__expf(v.x - m.x), __expf(v.y - m.y),
                   __expf(v.z - m.z), __expf(v.w - m.w)};
    ((float4*)ebuf)[(size_t)b * EP + e] = make_float4(ex[0], ex[1], ex[2], ex[3]);
    float* sp = ssum + ((size_t)b * N_NODES + tg) * HEADS;
    #pragma unroll
    for (int h = 0; h < 4; ++h)
        atomicAdd(sp + h, ex[h]);
}

// ---------- edge pass 3: weighted aggregation (head mean folded in) ----------
// wave per edge. grid: (ceil(EP/8), BATCH), block 256.
__global__ void edge_agg_kernel(const int* __restrict__ srcI,
                                const int* __restrict__ tgtI,
                                const float* __restrict__ ebuf,
                                const float* __restrict__ ssum,
                                const float* __restrict__ H,
                                float* __restrict__ agg) {
    const int b    = blockIdx.y;
    const int wave = threadIdx.x >> 5;
    const int lane = threadIdx.x & 31;
    const int e    = blockIdx.x * 8 + wave;
    if (e >= EP) return;
    int s, tg;
    if (e < N_EDGES) {
        s  = srcI[(size_t)b * N_EDGES + e];
        tg = tgtI[(size_t)b * N_EDGES + e];
    } else {
        s = tg = e - N_EDGES;
    }
    const float4 ex = ((const float4*)ebuf)[(size_t)b * EP + e];
    const float4 sv = ((const float4*)ssum)[(size_t)b * N_NODES + tg];
    const float al[4] = {0.25f * ex.x / sv.x, 0.25f * ex.y / sv.y,
                         0.25f * ex.z / sv.z, 0.25f * ex.w / sv.w};
    const float* hs = H + ((size_t)b * N_NODES + s) * HH;
    float4 acc = make_float4(0.f, 0.f, 0.f, 0.f);
    #pragma unroll
    for (int hd = 0; hd < 4; ++hd) {
        const float4 hv = ((const float4*)(hs + hd * HID))[lane];
        acc.x += al[hd] * hv.x;
        acc.y += al[hd] * hv.y;
        acc.z += al[hd] * hv.z;
        acc.w += al[hd] * hv.w;
    }
    float* ag = agg + ((size_t)b * N_NODES + tg) * HID + lane * 4;
    atomicAdd(ag + 0, acc.x);
    atomicAdd(ag + 1, acc.y);
    atomicAdd(ag + 2, acc.z);
    atomicAdd(ag + 3, acc.w);
}

// ---------- post: bias (+ LayerNorm + ReLU). wave per node ----------
__global__ void post_kernel(const float* __restrict__ agg,
                            const float* __restrict__ bias,
                            const float* __restrict__ lng,
                            const float* __restrict__ lnb,
                            float* __restrict__ out,
                            int do_ln) {
    const int b    = blockIdx.y;
    const int wave = threadIdx.x >> 5;
    const int lane = threadIdx.x & 31;
    const int n    = blockIdx.x * 8 + wave;
    if (n >= N_NODES) return;

    const float4 a4 = ((const float4*)(agg + ((size_t)b * N_NODES + n) * HID))[lane];
    const float4 bi = ((const float4*)bias)[lane];
    float x0 = a4.x + bi.x, x1 = a4.y + bi.y, x2 = a4.z + bi.z, x3 = a4.w + bi.w;

    if (do_ln) {
        float s = waveReduceSum(x0 + x1 + x2 + x3);
        const float mu = s * (1.0f / HID);
        const float d0 = x0 - mu, d1 = x1 - mu, d2 = x2 - mu, d3 = x3 - mu;
        float sq = waveReduceSum(d0 * d0 + d1 * d1 + d2 * d2 + d3 * d3);
        const float rs = rsqrtf(sq * (1.0f / HID) + LN_EPS);
        const float4 gv = ((const float4*)lng)[lane];
        const float4 bv = ((const float4*)lnb)[lane];
        x0 = fmaxf(d0 * rs * gv.x + bv.x, 0.f);
        x1 = fmaxf(d1 * rs * gv.y + bv.y, 0.f);
        x2 = fmaxf(d2 * rs * gv.z + bv.z, 0.f);
        x3 = fmaxf(d3 * rs * gv.w + bv.w, 0.f);
    }
    ((float4*)(out + ((size_t)b * N_NODES + n) * HID))[lane] = make_float4(x0, x1, x2, x3);
}

extern "C" void kernel_launch(void* const* d_in, const int* in_sizes, int n_in,
                              void* d_out, int out_size, void* d_ws, size_t ws_size,
                              hipStream_t stream) {
    (void)in_sizes; (void)n_in; (void)out_size; (void)ws_size;

    const float* x0   = (const float*)d_in[0];
    const int*   srcI = (const int*)d_in[1];
    const int*   tgtI = (const int*)d_in[2];
    const float* Wl[3]  = {(const float*)d_in[3], (const float*)d_in[7],  (const float*)d_in[11]};
    const float* Asl[3] = {(const float*)d_in[4], (const float*)d_in[8],  (const float*)d_in[12]};
    const float* Adl[3] = {(const float*)d_in[5], (const float*)d_in[9],  (const float*)d_in[13]};
    const float* Bl[3]  = {(const float*)d_in[6], (const float*)d_in[10], (const float*)d_in[14]};
    const float* lg[2]  = {(const float*)d_in[15], (const float*)d_in[17]};
    const float* lb[2]  = {(const float*)d_in[16], (const float*)d_in[18]};

    // workspace carve (f32 elements); all offsets 16B aligned
    float* ws   = (float*)d_ws;
    float* Hbuf = ws; ws += (size_t)BATCH * N_NODES * HH;      // 81.9 MB
    float* xbuf = ws; ws += (size_t)BATCH * N_NODES * HID;     // 20.5 MB
    float* esrc = ws; ws += (size_t)BATCH * N_NODES * HEADS;
    float* edst = ws; ws += (size_t)BATCH * N_NODES * HEADS;
    float* mmax = ws; ws += (size_t)BATCH * N_NODES * HEADS;
    float* ssum = ws; ws += (size_t)BATCH * N_NODES * HEADS;
    float* ebuf = ws; ws += (size_t)BATCH * EP * HEADS;        // 10.9 MB

    float* outp = (float*)d_out;

    for (int l = 0; l < 3; ++l) {
        const float* xin = (l == 0) ? x0 : xbuf;
        float* agg = (l == 2) ? outp : xbuf;   // input buffer is dead after GEMM

        gemm_wmma_f32<<<dim3(N_NODES / 16, BATCH), 256, 0, stream>>>(xin, Wl[l], Hbuf);
        init_kernel<<<dim3((N_NODES + 255) / 256, BATCH), 256, 0, stream>>>(mmax, ssum, agg);
        att_e_kernel<<<dim3((N_NODES * HEADS + 7) / 8, BATCH), 256, 0, stream>>>(
            Hbuf, Asl[l], Adl[l], esrc, edst);
        edge_max_kernel<<<dim3((EP + 255) / 256, BATCH), 256, 0, stream>>>(
            srcI, tgtI, esrc, edst, ebuf, mmax);
        edge_sum_kernel<<<dim3((EP + 255) / 256, BATCH), 256, 0, stream>>>(
            tgtI, mmax, ebuf, ssum);
        edge_agg_kernel<<<dim3((EP + 7) / 8, BATCH), 256, 0, stream>>>(
            srcI, tgtI, ebuf, ssum, Hbuf, agg);
        post_kernel<<<dim3((N_NODES + 7) / 8, BATCH), 256, 0, stream>>>(
            agg, Bl[l],
            (l < 2) ? lg[l] : Bl[l], (l < 2) ? lb[l] : Bl[l],
            agg, (l < 2) ? 1 : 0);
    }
}